// EdgeEnhancedGAT_39170101739642
// MI455X (gfx1250) — compile-verified
//
#include <hip/hip_runtime.h>

// Edge-Enhanced GAT for MI455X (gfx1250, wave32).
// Layer-1 projection GEMM uses V_WMMA_F32_16X16X4_F32 (fp32 WMMA, exact
// reference semantics) with a swizzled LDS staging of W1 so each B operand
// is one contiguous 8-byte ds_load (no VGPR repacking between WMMAs).
// Graph attention (segment softmax + scatter-add) uses float atomics; the
// edge MLP is folded into a single per-edge logit kernel via q = We@ae.

#define HEADS 4
#define HID 16
#define NEG_SLOPE 0.2f

typedef float v2f __attribute__((ext_vector_type(2)));
typedef float v8f __attribute__((ext_vector_type(8)));

__device__ __forceinline__ void atomicMaxF(float* addr, float val) {
  // Correct for any sign mix, given init to -inf.
  if (val >= 0.0f) atomicMax((int*)addr, __float_as_int(val));
  else             atomicMin((unsigned int*)addr, __float_as_uint(val));
}

__device__ __forceinline__ float lrelu(float a) {
  return a > 0.0f ? a : NEG_SLOPE * a;
}

// ---------------- init ----------------
__global__ void init_kernel(float* segmax1, float* segsum1, float* agg1,
                            float* segmax2, float* segsum2, float* agg2, int Nn) {
  int i = blockIdx.x * blockDim.x + threadIdx.x;
  if (i < Nn * 64) agg1[i] = 0.0f;
  if (i < Nn * 4) { segmax1[i] = -__builtin_inff(); segsum1[i] = 0.0f; }
  if (i < Nn * 2) agg2[i] = 0.0f;
  if (i < Nn)     { segmax2[i] = -__builtin_inff(); segsum2[i] = 0.0f; }
}

// ---------------- precompute q = We @ ae (per head) ----------------
__global__ void qprep_kernel(const float* __restrict__ We1, const float* __restrict__ ae1,
                             const float* __restrict__ We2, const float* __restrict__ ae2,
                             float* __restrict__ q) {
  if (blockIdx.x == 0 && threadIdx.x == 0) {
    for (int d = 0; d < 3; ++d)
      for (int h = 0; h < HEADS; ++h) {
        float s = 0.0f;
        for (int c = 0; c < HID; ++c)
          s += We1[d * (HEADS * HID) + h * HID + c] * ae1[h * HID + c];
        q[d * HEADS + h] = s;
      }
    for (int d = 0; d < 3; ++d) {
      float s = 0.0f;
      for (int c = 0; c < 2; ++c) s += We2[d * 2 + c] * ae2[c];
      q[12 + d] = s;
    }
  }
}

// ---------------- layer-1 GEMM: xs1 = x @ W1, fp32 WMMA ----------------
// One wave computes a 16x64 output tile; 8 waves per block.
// W1 (128x64 fp32 = 32KB) staged in LDS with a WMMA-friendly swizzle:
//   element (k,c) -> lds[ ((k>>2)*2 + ((k&3)>>1))*128 + c*2 + (k&1) ]
// so lane (half,lm) reads B operand {K=2*half+0, K=2*half+1} for column c
// as ONE contiguous, 8-byte-aligned float2.
__global__ __launch_bounds__(256) void gemm1_wmma_kernel(
    const float* __restrict__ x, const float* __restrict__ W1,
    float* __restrict__ xs1, int Nn) {
  __shared__ __align__(16) float ldsW[128 * 64];
  for (int i = threadIdx.x; i < 128 * 64; i += 256) {
    int k = i >> 6;
    int c = i & 63;
    int idx = (((k >> 2) * 2 + ((k & 3) >> 1)) * 64 + c) * 2 + (k & 1);
    ldsW[idx] = W1[i];
  }
  __syncthreads();

  const int wave = threadIdx.x >> 5;
  const int lane = threadIdx.x & 31;
  const int row0 = (blockIdx.x * 8 + wave) * 16;
  if (row0 >= Nn) return;                 // wave-uniform: EXEC stays all-1s
  const int half = lane >> 4;             // 0: lanes 0-15, 1: lanes 16-31
  const int lm = lane & 15;

  v8f acc0 = {}, acc1 = {}, acc2 = {}, acc3 = {};
  // A (16x4 fp32): lane half selects K pair {0,1} vs {2,3}, M = lane%16
  const float* xrow = x + (size_t)(row0 + lm) * 128 + 2 * half;
  __builtin_prefetch(xrow, 0, 0);
  // B base for this lane: kg*256 + half*128 + lm*2 ; col tile d adds d*32
  const float* wbase = &ldsW[half * 128 + lm * 2];

#pragma unroll 8
  for (int k0 = 0; k0 < 128; k0 += 4) {
    v2f a;
    a.x = xrow[k0 + 0];
    a.y = xrow[k0 + 1];
    const float* wk = wbase + (k0 >> 2) * 256;
    v2f b0 = *(const v2f*)(wk + 0);
    v2f b1 = *(const v2f*)(wk + 32);
    v2f b2 = *(const v2f*)(wk + 64);
    v2f b3 = *(const v2f*)(wk + 96);
    acc0 = __builtin_amdgcn_wmma_f32_16x16x4_f32(false, a, false, b0, (short)0, acc0, false, false);
    acc1 = __builtin_amdgcn_wmma_f32_16x16x4_f32(false, a, false, b1, (short)0, acc1, false, false);
    acc2 = __builtin_amdgcn_wmma_f32_16x16x4_f32(false, a, false, b2, (short)0, acc2, false, false);
    acc3 = __builtin_amdgcn_wmma_f32_16x16x4_f32(false, a, false, b3, (short)0, acc3, false, false);
  }

  // D (16x16 fp32): vgpr r, lane half -> M = r + half*8, N = lane%16
  float* outb = xs1 + (size_t)row0 * 64 + lm;
#pragma unroll
  for (int r = 0; r < 8; ++r) {
    float* orow = outb + (size_t)(r + half * 8) * 64;
    orow[0]  = acc0[r];
    orow[16] = acc1[r];
    orow[32] = acc2[r];
    orow[48] = acc3[r];
  }
}

// ---------------- per-node attention logits, layer 1 ----------------
__global__ void node_logits1_kernel(const float* __restrict__ xs1,
                                    const float* __restrict__ as1, const float* __restrict__ ad1,
                                    float* __restrict__ a_src1, float* __restrict__ a_dst1, int Nn) {
  int i = blockIdx.x * blockDim.x + threadIdx.x;
  if (i >= Nn * HEADS) return;
  int n = i >> 2, h = i & 3;
  const float* row = xs1 + (size_t)n * 64 + h * HID;
  float s = 0.0f, d = 0.0f;
#pragma unroll
  for (int c = 0; c < HID; ++c) {
    float v = row[c];
    s += v * as1[h * HID + c];
    d += v * ad1[h * HID + c];
  }
  a_src1[i] = s;
  a_dst1[i] = d;
}

// ---------------- edge MLP + edge logits (both layers) ----------------
__global__ void edge_kernel(const float* __restrict__ ef,
                            const float* __restrict__ Em1, const float* __restrict__ eb1,
                            const float* __restrict__ Em2, const float* __restrict__ eb2,
                            const float* __restrict__ q,
                            float* __restrict__ a_edge1, float* __restrict__ a_edge2, int Ee) {
  int e = blockIdx.x * blockDim.x + threadIdx.x;
  if (e >= Ee) return;
  float f0 = ef[(size_t)e * 3 + 0], f1 = ef[(size_t)e * 3 + 1], f2 = ef[(size_t)e * 3 + 2];
  float t[3], o[3];
#pragma unroll
  for (int d = 0; d < 3; ++d) {
    float v = f0 * Em1[0 * 3 + d] + f1 * Em1[1 * 3 + d] + f2 * Em1[2 * 3 + d] + eb1[d];
    t[d] = v > 0.0f ? v : 0.0f;
  }
#pragma unroll
  for (int d = 0; d < 3; ++d)
    o[d] = t[0] * Em2[0 * 3 + d] + t[1] * Em2[1 * 3 + d] + t[2] * Em2[2 * 3 + d] + eb2[d];
#pragma unroll
  for (int h = 0; h < HEADS; ++h)
    a_edge1[(size_t)e * HEADS + h] = o[0] * q[0 * HEADS + h] + o[1] * q[1 * HEADS + h] + o[2] * q[2 * HEADS + h];
  a_edge2[e] = o[0] * q[12] + o[1] * q[13] + o[2] * q[14];
}

// ---------------- layer-1 attention: logits + segment max ----------------
__global__ void alpha1_kernel(const int* __restrict__ srcI, const int* __restrict__ dstI,
                              const float* __restrict__ a_src1, const float* __restrict__ a_dst1,
                              const float* __restrict__ a_edge1,
                              float* __restrict__ alpha1, float* __restrict__ segmax1, int Ee) {
  int i = blockIdx.x * blockDim.x + threadIdx.x;
  if (i >= Ee * HEADS) return;
  int e = i >> 2, h = i & 3;
  int s = srcI[e], d = dstI[e];
  float a = a_src1[s * 4 + h] + a_dst1[d * 4 + h] + a_edge1[i];
  a = lrelu(a);
  alpha1[i] = a;
  atomicMaxF(&segmax1[d * 4 + h], a);
}

__global__ void exp1_kernel(const int* __restrict__ dstI,
                            const float* __restrict__ segmax1,
                            float* __restrict__ alpha1, float* __restrict__ segsum1, int Ee) {
  int i = blockIdx.x * blockDim.x + threadIdx.x;
  if (i >= Ee * HEADS) return;
  int e = i >> 2, h = i & 3;
  int d = dstI[e];
  float w = expf(alpha1[i] - segmax1[d * 4 + h]);
  alpha1[i] = w;
  atomicAdd(&segsum1[d * 4 + h], w);
}

__global__ void norm1_kernel(const int* __restrict__ dstI,
                             const float* __restrict__ segsum1,
                             float* __restrict__ alpha1, int Ee) {
  int i = blockIdx.x * blockDim.x + threadIdx.x;
  if (i >= Ee * HEADS) return;
  int e = i >> 2, h = i & 3;
  int d = dstI[e];
  alpha1[i] = alpha1[i] / (segsum1[d * 4 + h] + 1e-16f);
}

// ---------------- layer-1 weighted aggregation (scatter-add) ----------------
__global__ void agg1_kernel(const int* __restrict__ srcI, const int* __restrict__ dstI,
                            const float* __restrict__ xs1, const float* __restrict__ alpha1,
                            float* __restrict__ agg1, long long tot) {
  long long i = (long long)blockIdx.x * blockDim.x + threadIdx.x;
  if (i >= tot) return;
  int e = (int)(i >> 6);
  int j = (int)(i & 63);
  int h = j >> 4;
  int s = srcI[e], d = dstI[e];
  float v = xs1[(size_t)s * 64 + j] * alpha1[(size_t)e * 4 + h];
  atomicAdd(&agg1[(size_t)d * 64 + j], v);
}

// ---------------- bias + ELU (in place -> h) ----------------
__global__ void elu_bias1_kernel(float* __restrict__ agg1, const float* __restrict__ b1, int Nn) {
  int i = blockIdx.x * blockDim.x + threadIdx.x;
  if (i >= Nn * 64) return;
  float v = agg1[i] + b1[i & 63];
  agg1[i] = v > 0.0f ? v : (expf(v) - 1.0f);
}

// ---------------- layer-2 projection + node logits ----------------
__global__ void gemm2_kernel(const float* __restrict__ h, const float* __restrict__ W2,
                             const float* __restrict__ as2, const float* __restrict__ ad2,
                             float* __restrict__ xs2, float* __restrict__ a_src2,
                             float* __restrict__ a_dst2, int Nn) {
  int n = blockIdx.x * blockDim.x + threadIdx.x;
  if (n >= Nn) return;
  const float* hr = h + (size_t)n * 64;
  float o0 = 0.0f, o1 = 0.0f;
#pragma unroll 8
  for (int k = 0; k < 64; ++k) {
    float hv = hr[k];
    o0 += hv * W2[k * 2 + 0];
    o1 += hv * W2[k * 2 + 1];
  }
  xs2[n * 2 + 0] = o0;
  xs2[n * 2 + 1] = o1;
  a_src2[n] = o0 * as2[0] + o1 * as2[1];
  a_dst2[n] = o0 * ad2[0] + o1 * ad2[1];
}

// ---------------- layer-2 attention ----------------
__global__ void alpha2_kernel(const int* __restrict__ srcI, const int* __restrict__ dstI,
                              const float* __restrict__ a_src2, const float* __restrict__ a_dst2,
                              const float* __restrict__ a_edge2,
                              float* __restrict__ alpha2, float* __restrict__ segmax2, int Ee) {
  int e = blockIdx.x * blockDim.x + threadIdx.x;
  if (e >= Ee) return;
  int s = srcI[e], d = dstI[e];
  float a = lrelu(a_src2[s] + a_dst2[d] + a_edge2[e]);
  alpha2[e] = a;
  atomicMaxF(&segmax2[d], a);
}

__global__ void exp2_kernel(const int* __restrict__ dstI, const float* __restrict__ segmax2,
                            float* __restrict__ alpha2, float* __restrict__ segsum2, int Ee) {
  int e = blockIdx.x * blockDim.x + threadIdx.x;
  if (e >= Ee) return;
  int d = dstI[e];
  float w = expf(alpha2[e] - segmax2[d]);
  alpha2[e] = w;
  atomicAdd(&segsum2[d], w);
}

__global__ void agg2_kernel(const int* __restrict__ srcI, const int* __restrict__ dstI,
                            const float* __restrict__ xs2, const float* __restrict__ alpha2,
                            const float* __restrict__ segsum2, float* __restrict__ agg2, int Ee) {
  int e = blockIdx.x * blockDim.x + threadIdx.x;
  if (e >= Ee) return;
  int s = srcI[e], d = dstI[e];
  float coeff = alpha2[e] / (segsum2[d] + 1e-16f);
  atomicAdd(&agg2[d * 2 + 0], xs2[s * 2 + 0] * coeff);
  atomicAdd(&agg2[d * 2 + 1], xs2[s * 2 + 1] * coeff);
}

__global__ void final_kernel(const float* __restrict__ agg2, const float* __restrict__ b2,
                             float* __restrict__ out, int Nn) {
  int i = blockIdx.x * blockDim.x + threadIdx.x;
  if (i >= Nn * 2) return;
  out[i] = agg2[i] + b2[i & 1];
}

extern "C" void kernel_launch(void* const* d_in, const int* in_sizes, int n_in,
                              void* d_out, int out_size, void* d_ws, size_t ws_size,
                              hipStream_t stream) {
  const float* x   = (const float*)d_in[0];
  const int*   ei  = (const int*)d_in[1];
  const float* ef  = (const float*)d_in[2];
  const float* Em1 = (const float*)d_in[3];
  const float* eb1 = (const float*)d_in[4];
  const float* Em2 = (const float*)d_in[5];
  const float* eb2 = (const float*)d_in[6];
  const float* W1  = (const float*)d_in[7];
  const float* as1 = (const float*)d_in[8];
  const float* ad1 = (const float*)d_in[9];
  const float* We1 = (const float*)d_in[10];
  const float* ae1 = (const float*)d_in[11];
  const float* b1  = (const float*)d_in[12];
  const float* W2  = (const float*)d_in[13];
  const float* as2 = (const float*)d_in[14];
  const float* ad2 = (const float*)d_in[15];
  const float* We2 = (const float*)d_in[16];
  const float* ae2 = (const float*)d_in[17];
  const float* b2  = (const float*)d_in[18];

  const int Nn = in_sizes[0] / 128;   // 100000
  const int Ee = in_sizes[2] / 3;     // 1600000
  const int* srcI = ei;
  const int* dstI = ei + Ee;

  // workspace layout (floats)
  float* ws = (float*)d_ws;
  size_t o = 0;
  float* xs1     = ws + o; o += (size_t)Nn * 64;
  float* a_src1  = ws + o; o += (size_t)Nn * 4;
  float* a_dst1  = ws + o; o += (size_t)Nn * 4;
  float* a_edge1 = ws + o; o += (size_t)Ee * 4;
  float* a_edge2 = ws + o; o += (size_t)Ee;
  float* alpha1  = ws + o; o += (size_t)Ee * 4;
  float* segmax1 = ws + o; o += (size_t)Nn * 4;
  float* segsum1 = ws + o; o += (size_t)Nn * 4;
  float* agg1    = ws + o; o += (size_t)Nn * 64;
  float* xs2     = ws + o; o += (size_t)Nn * 2;
  float* a_src2  = ws + o; o += (size_t)Nn;
  float* a_dst2  = ws + o; o += (size_t)Nn;
  float* alpha2  = ws + o; o += (size_t)Ee;
  float* segmax2 = ws + o; o += (size_t)Nn;
  float* segsum2 = ws + o; o += (size_t)Nn;
  float* agg2    = ws + o; o += (size_t)Nn * 2;
  float* q       = ws + o; o += 16;

  const int B = 256;
  int gN64 = (Nn * 64 + B - 1) / B;
  int gNH  = (Nn * 4 + B - 1) / B;
  int gN   = (Nn + B - 1) / B;
  int gN2  = (Nn * 2 + B - 1) / B;
  int gE   = (Ee + B - 1) / B;
  int gEH  = (Ee * 4 + B - 1) / B;
  long long totAgg1 = (long long)Ee * 64;
  int gE64 = (int)((totAgg1 + B - 1) / B);
  int gGemm = ((Nn + 15) / 16 + 7) / 8;   // 8 waves/block, 16 rows/wave

  init_kernel<<<gN64, B, 0, stream>>>(segmax1, segsum1, agg1, segmax2, segsum2, agg2, Nn);
  qprep_kernel<<<1, 32, 0, stream>>>(We1, ae1, We2, ae2, q);
  gemm1_wmma_kernel<<<gGemm, B, 0, stream>>>(x, W1, xs1, Nn);
  node_logits1_kernel<<<gNH, B, 0, stream>>>(xs1, as1, ad1, a_src1, a_dst1, Nn);
  edge_kernel<<<gE, B, 0, stream>>>(ef, Em1, eb1, Em2, eb2, q, a_edge1, a_edge2, Ee);
  alpha1_kernel<<<gEH, B, 0, stream>>>(srcI, dstI, a_src1, a_dst1, a_edge1, alpha1, segmax1, Ee);
  exp1_kernel<<<gEH, B, 0, stream>>>(dstI, segmax1, alpha1, segsum1, Ee);
  norm1_kernel<<<gEH, B, 0, stream>>>(dstI, segsum1, alpha1, Ee);
  agg1_kernel<<<gE64, B, 0, stream>>>(srcI, dstI, xs1, alpha1, agg1, totAgg1);
  elu_bias1_kernel<<<gN64, B, 0, stream>>>(agg1, b1, Nn);
  gemm2_kernel<<<gN, B, 0, stream>>>(agg1, W2, as2, ad2, xs2, a_src2, a_dst2, Nn);
  alpha2_kernel<<<gE, B, 0, stream>>>(srcI, dstI, a_src2, a_dst2, a_edge2, alpha2, segmax2, Ee);
  exp2_kernel<<<gE, B, 0, stream>>>(dstI, segmax2, alpha2, segsum2, Ee);
  agg2_kernel<<<gE, B, 0, stream>>>(srcI, dstI, xs2, alpha2, segsum2, agg2, Ee);
  final_kernel<<<gN2, B, 0, stream>>>(agg2, b2, (float*)d_out, Nn);
}